// PoseVocab_65987877536088
// MI455X (gfx1250) — compile-verified
//
#include <hip/hip_runtime.h>
#include <hip/hip_bf16.h>
#include <math.h>

// Problem constants (match the reference)
#define B_ 4
#define N_ 16384
#define J_ 24
#define P_ 512
#define K_ 10
#define L_ 128
#define C_ 16
#define LC_ 2048            // L_*C_
#define JG_ 4               // j's per interp block
#define TN_ 128             // query points per interp block
#define ITHREADS 384        // interp block size (12 waves of 32)

typedef float v2f __attribute__((ext_vector_type(2)));
typedef float v4f __attribute__((ext_vector_type(4)));
typedef float v8f __attribute__((ext_vector_type(8)));
typedef int   v4i __attribute__((ext_vector_type(4)));

#define AS1 __attribute__((address_space(1)))
#define AS3 __attribute__((address_space(3)))

// gfx1250 async global->LDS copy path (ASYNCcnt-tracked), guarded so both the
// host pass and older toolchains fall back to the plain VGPR-bounce copy.
#if defined(__gfx1250__) && __has_builtin(__builtin_amdgcn_global_load_async_to_lds_b128) && \
    __has_builtin(__builtin_amdgcn_s_wait_asynccnt)
#define USE_ASYNC_LDS 1
#else
#define USE_ASYNC_LDS 0
#endif

// ---------------------------------------------------------------------------
// Kernel 1a: pose_dist[b,j,p] = |dot4(query_poses[b,j,:], pose_points[j,p,:])|
// One wave (32 lanes) per j. Per 16-wide p-tile: D(16x16) = A(16x4) x B(4x16)
// via V_WMMA_F32_16X16X4_F32. M rows = b (rows 4..15 zero-padded).
// A layout (32-bit 16x4): lanes 0-15 -> {K=0,K=1} in v[0],v[1]; lanes 16-31 -> {K=2,K=3}.
// B layout (4x16, row striped across lanes): v[0]: K=0 (lanes<16) / K=2 (lanes>=16);
//                                            v[1]: K=1 / K=3.
// D layout: VGPR r -> M = r + 8*(lane>=16), N = lane&15.
// ---------------------------------------------------------------------------
__global__ void k_pose_dist_wmma(const float* __restrict__ qposes,   // (B,J,4)
                                 const float* __restrict__ ppoints,  // (J,P,4)
                                 float* __restrict__ dist)           // (B,J,P)
{
    const int j     = blockIdx.x;
    const int lane  = threadIdx.x;        // 0..31
    const int m     = lane & 15;          // M row / N col
    const int khalf = lane >> 4;          // 0 or 1

    v2f a;
    if (m < B_) {
        const float* qp = qposes + ((size_t)m * J_ + j) * 4 + 2 * khalf;
        a.x = qp[0];
        a.y = qp[1];
    } else {
        a.x = 0.0f;
        a.y = 0.0f;
    }

    for (int p0 = 0; p0 < P_; p0 += 16) {
        const int p = p0 + m;             // N column -> p index
        const float* pp = ppoints + (((size_t)j * P_ + p) * 4) + 2 * khalf;
        v2f bm;
        bm.x = pp[0];
        bm.y = pp[1];

        v8f c = {};
        c = __builtin_amdgcn_wmma_f32_16x16x4_f32(
                /*neg_a=*/false, a, /*neg_b=*/false, bm,
                /*c_mod=*/(short)0, c, /*reuse_a=*/false, /*reuse_b=*/false);

        if (khalf == 0) {                 // rows M=0..7 live in lanes 0..15
#pragma unroll
            for (int r = 0; r < 4; ++r) { // only M=0..3 are valid b rows
                dist[((size_t)r * J_ + j) * P_ + p] = fabsf(c[r]);
            }
        }
    }
}

// ---------------------------------------------------------------------------
// Kernel 1b: per-(b,j) top-K over P=512 with lax.top_k semantics
// (descending, ties -> lower index wins via strict '>'), then L1-normalize.
// 96 independent problems -> one thread each; cost is negligible.
// ---------------------------------------------------------------------------
__global__ void k_topk(const float* __restrict__ dist,  // (B,J,P)
                       float* __restrict__ tw,          // (B,J,K)
                       int* __restrict__ tid)           // (B,J,K)
{
    const int g = blockIdx.x * blockDim.x + threadIdx.x;
    if (g >= B_ * J_) return;
    const float* d = dist + (size_t)g * P_;

    int   ids[K_];
    float vs[K_];
#pragma unroll
    for (int k = 0; k < K_; ++k) {
        float bv = -1.0f;   // dist >= 0 always
        int   bi = 0;
        for (int p = 0; p < P_; ++p) {
            bool taken = false;
#pragma unroll
            for (int q = 0; q < K_; ++q)
                if (q < k && ids[q] == p) taken = true;
            const float v = d[p];
            if (!taken && v > bv) { bv = v; bi = p; }
        }
        ids[k] = bi;
        vs[k]  = bv;
    }
    float s = 0.0f;
#pragma unroll
    for (int k = 0; k < K_; ++k) s += fabsf(vs[k]);
    s = fmaxf(s, 1e-16f);
#pragma unroll
    for (int k = 0; k < K_; ++k) {
        tw[(size_t)g * K_ + k]  = vs[k] / s;
        tid[(size_t)g * K_ + k] = ids[k];
    }
}

// ---------------------------------------------------------------------------
// Kernel 2: comb[axis,b,j,l,c] = sum_k tw[b,j,k] * lines_axis[j, tid[b,j,k], l, c]
// Each block owns one (axis,b,j) slice chunk; w/id loads are block-uniform
// (scalar), line rows are 8KB coalesced gathers. ~28 MB total traffic.
// ---------------------------------------------------------------------------
__global__ void k_comb(const float* __restrict__ fx,
                       const float* __restrict__ fy,
                       const float* __restrict__ fz,
                       const float* __restrict__ tw,
                       const int* __restrict__ tid,
                       float* __restrict__ comb)        // (3,B,J,L,C)
{
    const int gid  = blockIdx.x * 256 + threadIdx.x;
    const int e    = gid & (LC_ - 1);
    int rest       = gid >> 11;                  // LC_ == 2048 == 1<<11
    const int j    = rest % J_;  rest /= J_;
    const int b    = rest % B_;
    const int axis = rest / B_;

    const float* lines = (axis == 0) ? fx : ((axis == 1) ? fy : fz);
    const float* w  = tw  + ((size_t)b * J_ + j) * K_;
    const int*   id = tid + ((size_t)b * J_ + j) * K_;

    float acc = 0.0f;
#pragma unroll
    for (int k = 0; k < K_; ++k) {
        acc += w[k] * lines[(((size_t)j * P_ + id[k]) << 11) + e];
    }
    comb[((((size_t)axis * B_ + b) * J_) + j) * LC_ + e] = acc;
}

// ---------------------------------------------------------------------------
// Kernel 3: streaming interpolation (the bandwidth-bound phase).
// Block = (b, group of JG_=4 j's, TN_=128 query points). Stages its
// 3*JG_*8KB = 96KB comb slice into LDS via GLOBAL_LOAD_ASYNC_TO_LDS_B128
// (ASYNCcnt path; VGPR-bounce fallback), then each thread produces one float4
// (4 c's) per (n, j, axis): 2 LDS b128 loads -> lerp -> one 128-bit
// NON-TEMPORAL global store (output never re-read; 302 MB stream).
// Thread map: cq = t%4, axis = (t/4)%3, jl = (t/12)%4, nsub = t/48 (8 n/iter).
// ---------------------------------------------------------------------------
__global__ void k_interp(const float* __restrict__ qpts,   // (B,N,3) in [0,1)
                         const float* __restrict__ comb,   // (3,B,J,L,C)
                         float* __restrict__ out)          // (B,N,J,3,C)
{
    extern __shared__ float sm[];
    float* scomb = sm;                    // [3][JG_][L_][C_] = 24576 floats
    float* sq    = sm + 3 * JG_ * LC_;    // [TN_][3]

    const int tid3 = threadIdx.x;
    const int b    = blockIdx.z;
    const int jg   = blockIdx.y;
    const int n0   = blockIdx.x * TN_;

    // ---- Stage comb slice into LDS (b128 granularity: 6144 transfers) ----
    const v4f* gcomb4 = (const v4f*)comb;
    const int  CE4    = 3 * JG_ * (LC_ / 4);    // 6144 float4s
#if USE_ASYNC_LDS
    {
        AS3 char* sbase = (AS3 char*)scomb;
        for (int idx = tid3; idx < CE4; idx += ITHREADS) {
            const int e4   = idx & (LC_ / 4 - 1);          // 0..511
            const int jl   = (idx >> 9) & (JG_ - 1);
            const int axis = idx >> 11;                    // JG_*512 == 2048
            const v4f* src = gcomb4 +
                ((((size_t)axis * B_ + b) * J_) + (jg * JG_ + jl)) * (LC_ / 4) + e4;
            __builtin_amdgcn_global_load_async_to_lds_b128(
                (AS1 v4i*)src, (AS3 v4i*)(sbase + (size_t)idx * 16),
                /*imm offset=*/0, /*cpol=*/0);
        }
    }
#else
    {
        v4f* scomb4 = (v4f*)scomb;
        for (int idx = tid3; idx < CE4; idx += ITHREADS) {
            const int e4   = idx & (LC_ / 4 - 1);
            const int jl   = (idx >> 9) & (JG_ - 1);
            const int axis = idx >> 11;
            scomb4[idx] = gcomb4[((((size_t)axis * B_ + b) * J_) + (jg * JG_ + jl)) * (LC_ / 4) + e4];
        }
    }
#endif
    // Stage this block's query points (contiguous slab of (N,3))
    const size_t qbase = ((size_t)b * N_ + n0) * 3;
    for (int idx = tid3; idx < TN_ * 3; idx += ITHREADS) sq[idx] = qpts[qbase + idx];

#if USE_ASYNC_LDS
    __builtin_amdgcn_s_wait_asynccnt(0);   // wave's async->LDS copies complete
#endif
    __syncthreads();                       // cross-wave visibility

    const int nsub = tid3 / 48;
    const int r    = tid3 % 48;
    const int jl   = r / 12;
    const int axis = (r / 4) % 3;
    const int cq   = r & 3;
    const int j    = jg * JG_ + jl;

    const v4f* srow4 = (const v4f*)(scomb + ((axis * JG_ + jl) * L_) * C_);
    v4f* out4 = (v4f*)out;

#pragma unroll 4
    for (int it = 0; it < TN_ / 8; ++it) {
        const int nn = it * 8 + nsub;     // local point index
        // normalize [0,1) -> [-1,1] -> grid coord: iy = q * (L-1) exactly
        const float qv  = sq[nn * 3 + axis];
        const float iy  = qv * (float)(L_ - 1);
        const float i0f = floorf(iy);
        const float t   = iy - i0f;
        int i0 = (int)i0f;
        i0 = i0 < 0 ? 0 : (i0 > L_ - 1 ? L_ - 1 : i0);
        const int i1 = (i0 + 1 > L_ - 1) ? (L_ - 1) : (i0 + 1);

        const v4f f0 = srow4[i0 * 4 + cq];   // ds_load_b128
        const v4f f1 = srow4[i1 * 4 + cq];
        const float u = 1.0f - t;
        v4f o;
        o.x = f0.x * u + f1.x * t;
        o.y = f0.y * u + f1.y * t;
        o.z = f0.z * u + f1.z * t;
        o.w = f0.w * u + f1.w * t;

        const size_t oi = ((((size_t)b * N_ + (n0 + nn)) * J_ + j) * 12) + axis * 4 + cq;
        __builtin_nontemporal_store(o, &out4[oi]);   // 128-bit NT streaming store
    }
}

// ---------------------------------------------------------------------------
// Host launcher
// ---------------------------------------------------------------------------
extern "C" void kernel_launch(void* const* d_in, const int* in_sizes, int n_in,
                              void* d_out, int out_size, void* d_ws, size_t ws_size,
                              hipStream_t stream) {
    const float* qpts   = (const float*)d_in[0];  // (B,N,3)
    const float* qposes = (const float*)d_in[1];  // (B,J,4)
    const float* ppts   = (const float*)d_in[2];  // (J,P,4)
    const float* fx     = (const float*)d_in[3];  // (J,P,L,C)
    const float* fy     = (const float*)d_in[4];
    const float* fz     = (const float*)d_in[5];
    float* out = (float*)d_out;

    // Workspace layout (≈2.6 MB total)
    char* ws = (char*)d_ws;
    float* dist = (float*)(ws);                               // B*J*P   = 49152 f
    float* tw   = (float*)(ws + (size_t)B_ * J_ * P_ * 4);    // B*J*K   = 960 f
    int*   tid  = (int*)((char*)tw + (size_t)B_ * J_ * K_ * 4);
    size_t comb_off = (((size_t)B_ * J_ * P_ * 4) + ((size_t)B_ * J_ * K_ * 8) + 255) & ~(size_t)255;
    float* comb = (float*)(ws + comb_off);                    // 3*B*J*L*C = 589824 f

    // 1a) WMMA pose-distance routing matmul: one wave per j
    k_pose_dist_wmma<<<J_, 32, 0, stream>>>(qposes, ppts, dist);

    // 1b) top-K + normalize: 96 independent tiny problems
    k_topk<<<1, 128, 0, stream>>>(dist, tw, tid);

    // 2) weighted gather-combine into the 2.25 MB comb table
    {
        const int total  = 3 * B_ * J_ * LC_;                 // 589824
        k_comb<<<total / 256, 256, 0, stream>>>(fx, fy, fz, tw, tid, comb);
    }

    // 3) streaming LDS-staged interpolation (write-bandwidth bound: 302 MB)
    {
        const size_t smem = (size_t)(3 * JG_ * LC_ + TN_ * 3) * sizeof(float); // ~97.5 KB
        (void)hipFuncSetAttribute((const void*)k_interp,
                                  hipFuncAttributeMaxDynamicSharedMemorySize, (int)smem);
        dim3 grid(N_ / TN_, J_ / JG_, B_);
        k_interp<<<grid, ITHREADS, smem, stream>>>(qpts, comb, out);
    }
}